// BaseSolver_18760417149386
// MI455X (gfx1250) — compile-verified
//
#include <hip/hip_runtime.h>
#include <stdint.h>

// ---------------- LBM / D2Q9 constants ----------------
#define NXg 2048
#define NYg 2048
#define TS  32           // output tile edge
#define EXT 34           // tile + 1-cell halo on each side

constexpr float TAU   = 0.6f;
constexpr float OMEGA = 1.0f / TAU;

constexpr int   cEX[9]  = {0, 1, 0, -1,  0, 1, -1, -1,  1};
constexpr int   cEY[9]  = {0, 0, 1,  0, -1, 1,  1, -1, -1};
constexpr int   cOPP[9] = {0, 3, 4,  1,  2, 7,  8,  5,  6};
constexpr float cW[9]   = {4.0f/9.0f,
                           1.0f/9.0f, 1.0f/9.0f, 1.0f/9.0f, 1.0f/9.0f,
                           1.0f/36.0f, 1.0f/36.0f, 1.0f/36.0f, 1.0f/36.0f};

typedef __attribute__((ext_vector_type(2))) float v2f;
typedef __attribute__((ext_vector_type(4))) float v4f;
typedef __attribute__((ext_vector_type(8))) float v8f;

// Branchless B-matrix element for the WMMA moment reduction.
// Columns: n==0 -> e_x[q], n==1 -> e_y[q], n==2 -> 1 (density), else 0.
// q is a compile-time literal at all call sites.
__device__ __forceinline__ float bval(int q, int n) {
  float r = (n == 0) ? (float)cEX[q] : 0.0f;
  r       = (n == 1) ? (float)cEY[q] : r;
  r       = (n == 2) ? 1.0f          : r;
  return r;
}

__global__ __launch_bounds__(256)
void lbm_step_kernel(const float* __restrict__ f,
                     const float* __restrict__ rho,
                     const float* __restrict__ u,
                     const unsigned char* __restrict__ mask,
                     float* __restrict__ out)
{
  __shared__ float         s_fstar[EXT * EXT * 9];   // 41,616 B post-collision tile
  __shared__ unsigned char s_mask[TS * TS];          // interior obstacle flags

  const int x0  = blockIdx.x * TS;
  const int y0  = blockIdx.y * TS;
  const int tid = threadIdx.x;

  // ---------------- Phase 1: collide (BGK) into LDS over haloed tile ----------------
  for (int i = tid; i < EXT * EXT; i += 256) {
    const int exi = i / EXT;
    const int eyi = i - exi * EXT;
    const int gx  = (x0 + exi - 1) & (NXg - 1);      // periodic wrap
    const int gy  = (y0 + eyi - 1) & (NYg - 1);
    const unsigned c = (unsigned)gx * NYg + (unsigned)gy;   // 32-bit addressing

    const float r   = rho[c];
    const v2f   uv  = *(const v2f*)(u + 2u * c);            // 8B-aligned b64 load
    const float ux  = uv.x, uy = uv.y;
    const float usq = ux * ux + uy * uy;

    const float* fc = f + 9u * c;
    float*       sf = s_fstar + i * 9;
#pragma unroll
    for (int q = 0; q < 9; ++q) {
      const float eu  = (float)cEX[q] * ux + (float)cEY[q] * uy;
      const float feq = cW[q] * r * (1.0f + 3.0f * eu + 4.5f * eu * eu - 1.5f * usq);
      const float fv  = fc[q];
      sf[q] = fv - (fv - feq) * OMEGA;
    }
    if (exi >= 1 && exi <= TS && eyi >= 1 && eyi <= TS) {
      s_mask[(exi - 1) * TS + (eyi - 1)] = mask[c];
    }
  }
  __syncthreads();

  // ---------------- Phase 2: stream + bounce-back + WMMA moment lift ----------------
  const int wave     = tid >> 5;
  const int lane     = tid & 31;
  const int m        = lane & 15;   // A row / B,D column index
  const int halfsel  = lane >> 4;   // 0: K={0,1}, 1: K={2,3}  (f32 A/B layout)
  const int halfbase = lane & 16;

  // B operands are lane-invariant across all rows/groups/chunks: hoist once.
  const float bx0 = halfsel ? bval(2, m) : bval(0, m);
  const float by0 = halfsel ? bval(3, m) : bval(1, m);
  const float bx1 = halfsel ? bval(6, m) : bval(4, m);
  const float by1 = halfsel ? bval(7, m) : bval(5, m);
  const float bx2 = halfsel ? 0.0f       : bval(8, m);

#pragma unroll
  for (int rr = 0; rr < 4; ++rr) {
    const int tx  = wave * 4 + rr;      // tile row handled by this wave
    const int exi = tx + 1;
    const unsigned rowbase = ((unsigned)(x0 + tx) * NYg + (unsigned)y0) * 12u;
    const bool solid  = (s_mask[tx * TS + lane] != 0);
    const int  base_c = (exi * EXT + (lane + 1)) * 9;   // this cell in s_fstar

    // Gather: select the *address offset* (both constants) -> 1 ds_load per q.
    float fn[9];
#pragma unroll
    for (int q = 0; q < 9; ++q) {
      // streamed: f_star[x - e_x, y - e_y][q]  == base_c + (q - 9*(34*ex + ey))
      const int offS = q - 9 * (EXT * cEX[q] + cEY[q]);
      const int off  = solid ? cOPP[q] : offS;          // bounce-back vs streamed
      fn[q] = s_fstar[base_c + off];
    }

    // write f_new: 48B/cell record is 16B aligned -> b128+b128+b32.
    // Output is write-once/never-read: non-temporal keeps the 192MB L2 free
    // for the inter-tile halo reuse of the inputs.
    const unsigned ob = rowbase + 12u * (unsigned)lane;
    v4f lo = {fn[0], fn[1], fn[2], fn[3]};
    v4f hi = {fn[4], fn[5], fn[6], fn[7]};
    __builtin_nontemporal_store(lo, (v4f*)(out + ob));
    __builtin_nontemporal_store(hi, (v4f*)(out + ob + 4));
    __builtin_nontemporal_store(fn[8], out + ob + 8);

    // Moment lift per 16-cell group: D[m][0..2] = (mom_x, mom_y, rho), via
    // 3 chained V_WMMA_F32_16X16X4_F32 (K = 9 zero-padded to 12).
#pragma unroll
    for (int h = 0; h < 2; ++h) {
      const int src = h * 16 + m;       // lane owning cell m of this group
      v8f acc = {};

      { // chunk 0: q = 0..3
        const float s0 = __shfl(fn[0], src, 32);
        const float s1 = __shfl(fn[1], src, 32);
        const float s2 = __shfl(fn[2], src, 32);
        const float s3 = __shfl(fn[3], src, 32);
        v2f a, b;
        a.x = halfsel ? s2 : s0;
        a.y = halfsel ? s3 : s1;
        b.x = bx0;
        b.y = by0;
        acc = __builtin_amdgcn_wmma_f32_16x16x4_f32(false, a, false, b,
                                                    (short)0, acc, false, false);
      }
      { // chunk 1: q = 4..7
        const float s4 = __shfl(fn[4], src, 32);
        const float s5 = __shfl(fn[5], src, 32);
        const float s6 = __shfl(fn[6], src, 32);
        const float s7 = __shfl(fn[7], src, 32);
        v2f a, b;
        a.x = halfsel ? s6 : s4;
        a.y = halfsel ? s7 : s5;
        b.x = bx1;
        b.y = by1;
        acc = __builtin_amdgcn_wmma_f32_16x16x4_f32(false, a, false, b,
                                                    (short)0, acc, false, false);
      }
      { // chunk 2: q = 8, K = 9..11 zero-padded
        const float s8 = __shfl(fn[8], src, 32);
        v2f a, b;
        a.x = halfsel ? 0.0f : s8;
        a.y = 0.0f;
        b.x = bx2;
        b.y = 0.0f;
        acc = __builtin_amdgcn_wmma_f32_16x16x4_f32(false, a, false, b,
                                                    (short)0, acc, false, false);
      }

      // rho per accumulator register, fetched cross-lane (source lane m==2 is
      // always active), with full EXEC:
      float rv[8];
#pragma unroll
      for (int r8 = 0; r8 < 8; ++r8) {
        rv[r8] = __shfl(acc[r8], halfbase + 2, 32);
      }

      // Single divergent region: lanes n==0/1/2 store u_x/u_y/rho for 8 cells.
      // u = mom * rcp(rho): v_rcp_f32 (+mul) instead of the ~12-op IEEE divide;
      // ~2 ulp relative error, far below validation tolerance.
      if (m < 3) {
        const unsigned gb = rowbase + (unsigned)(h * 16 + (halfbase ? 8 : 0)) * 12u;
        const unsigned co = (m == 2) ? 9u : (10u + (unsigned)m);
#pragma unroll
        for (int r8 = 0; r8 < 8; ++r8) {
          const float v   = acc[r8];
          const float val = (m == 2) ? v : v * __builtin_amdgcn_rcpf(rv[r8]);
          __builtin_nontemporal_store(val, out + gb + (unsigned)r8 * 12u + co);
        }
      }
    }
  }
}

extern "C" void kernel_launch(void* const* d_in, const int* in_sizes, int n_in,
                              void* d_out, int out_size, void* d_ws, size_t ws_size,
                              hipStream_t stream) {
  (void)in_sizes; (void)n_in; (void)out_size; (void)d_ws; (void)ws_size;
  const float*         f    = (const float*)d_in[0];
  const float*         rho  = (const float*)d_in[1];
  const float*         u    = (const float*)d_in[2];
  const unsigned char* mask = (const unsigned char*)d_in[3];  // jnp bool_: 1 byte/elem
  float* out = (float*)d_out;

  dim3 grid(NXg / TS, NYg / TS);
  lbm_step_kernel<<<grid, 256, 0, stream>>>(f, rho, u, mask, out);
}